// VectorQuantisizer_32547262169614
// MI455X (gfx1250) — compile-verified
//
#include <hip/hip_runtime.h>
#include <hip/hip_bf16.h>
#include <stdint.h>

// ---------------- problem constants ----------------
#define BATCH    64
#define DIM      32
#define HW       16384          // 128*128
#define NCODE    512
#define NVEC     1048576L       // BATCH*HW
#define LOSS_OFF 33554432L
#define DISC_OFF 33554433L
#define DISC_N   536870912L     // BATCH*NCODE*HW
// one-hot region starts at int offset DISC_OFF (byte offset % 16 == 4):
#define HEAD_N   3L
#define BODY4_N  134217727L     // (DISC_N - 3) / 4

typedef __attribute__((ext_vector_type(2))) float        v2f;
typedef __attribute__((ext_vector_type(8))) float        v8f;
typedef __attribute__((ext_vector_type(4))) unsigned int u32x4;
typedef __attribute__((ext_vector_type(4))) int          i32x4;
typedef __attribute__((ext_vector_type(8))) int          i32x8;

// ---------------- kernel 1: zero-fill one-hot + loss slot (NT b128) ----------------
__global__ __launch_bounds__(256) void vq_fill(float* __restrict__ out) {
    int* disc = ((int*)out) + DISC_OFF;
    u32x4* body = (u32x4*)(disc + HEAD_N);          // 16B aligned
    long i = (long)blockIdx.x * blockDim.x + threadIdx.x;
    const long stride = (long)gridDim.x * blockDim.x;
    const u32x4 z = { 0u, 0u, 0u, 0u };
    for (; i < BODY4_N; i += stride)
        __builtin_nontemporal_store(z, body + i);   // global_store_b128 th:NT
    if (blockIdx.x == 0 && threadIdx.x == 0) {
        out[LOSS_OFF] = 0.0f;                       // atomic target must start at 0
        disc[0] = 0; disc[1] = 0; disc[2] = 0;      // head
        disc[HEAD_N + BODY4_N * 4] = 0;             // tail
    }
}

// ---------------- kernel 2: WMMA distance/argmin core ----------------
// 256 threads = 8 waves; each wave owns 16 consecutive vectors (tiles never
// cross a batch image since HW % 16 == 0). Codebook (64 KB) staged into LDS
// by the Tensor Data Mover (one descriptor, wave 0 issues, TENSORcnt wait).
__global__ __launch_bounds__(256, 2) void vq_main(const float* __restrict__ x,
                                                  const float* __restrict__ emb,
                                                  float* __restrict__ qout,
                                                  float* __restrict__ loss,
                                                  int* __restrict__ idx_ws) {
    __shared__ float embLDS[NCODE * DIM];   // 64 KB (of 320 KB/WGP)
    __shared__ float e2LDS[NCODE];          // 2 KB
    __shared__ int   idxLDS[8 * 16];        // per-wave argmin results

    const int tid = threadIdx.x;

    // ---- TDM: DMA the whole 64 KB codebook into LDS with one descriptor ----
    if (tid < 32) {                                   // wave 0 issues (per-wave op)
        const uint64_t ga      = (uint64_t)(uintptr_t)emb;
        const uint32_t lds_off = (uint32_t)(uintptr_t)(void*)embLDS;
        // D# group0: count=1 | lds_addr | global_addr[56:0] | type=2
        u32x4 g0 = { 1u, lds_off, (uint32_t)ga,
                     (uint32_t)((ga >> 32) & 0x1FFFFFFull) | (2u << 30) };
        // D# group1: data_size=4B; tensor_dim0=16384, tensor_dim1=1;
        //            tile_dim0=16384, tile_dim1=1; dim0_stride=16384
        i32x8 g1 = { 0x00020000, 0x40000000, 0x00010000, 0x40000000,
                     1, 16384, 0, 0 };
        i32x4 gz = { 0, 0, 0, 0 };
#if __clang_major__ >= 23
        i32x8 gz8 = { 0, 0, 0, 0, 0, 0, 0, 0 };
        __builtin_amdgcn_tensor_load_to_lds(g0, g1, gz, gz, gz8, 0);
#else
        __builtin_amdgcn_tensor_load_to_lds(g0, g1, gz, gz, 0);
#endif
        __builtin_amdgcn_s_wait_tensorcnt(0);         // wave0 sees its own TDM done
    }
    __syncthreads();                                  // publish LDS to all waves

    // per-code squared norms
    for (int c = tid; c < NCODE; c += 256) {
        float s = 0.0f;
        #pragma unroll
        for (int d = 0; d < DIM; ++d) { float v = embLDS[c * DIM + d]; s += v * v; }
        e2LDS[c] = s;
    }
    __syncthreads();

    const int wave = tid >> 5;
    const int lane = tid & 31;
    const int m    = lane & 15;      // row within tile / code column within block
    const int half = lane >> 4;      // 0: K=+0,+1  1: K=+2,+3 (ISA A/B layout)

    const long nbase = (long)blockIdx.x * 128 + (long)wave * 16;
    const int  b   = (int)(nbase >> 14);
    const int  hw0 = (int)(nbase & (HW - 1));

    // ---- load A fragments: full 16x32 f32 tile in 16 VGPRs per lane (NT, read-once) ----
    const float* xb = x + (long)b * DIM * HW + hw0 + m;
    v2f a[8];
    #pragma unroll
    for (int s = 0; s < 8; ++s) {
        const int d0 = 4 * s + 2 * half;
        a[s].x = __builtin_nontemporal_load(xb + (long)d0 * HW);
        a[s].y = __builtin_nontemporal_load(xb + (long)(d0 + 1) * HW);
    }

    // ---- sweep 32 code blocks of 16; argmin of e2[c] - 2*dot (x2 const per row) ----
    float mind[8]; int mini[8];
    #pragma unroll
    for (int v = 0; v < 8; ++v) { mind[v] = 3.4e38f; mini[v] = 0; }

    for (int j = 0; j < 32; ++j) {
        const int c0 = j * 16;
        const float* ecol = &embLDS[(c0 + m) * DIM + 2 * half];
        v8f acc = {0.f, 0.f, 0.f, 0.f, 0.f, 0.f, 0.f, 0.f};
        #pragma unroll
        for (int s = 0; s < 8; ++s) {
            v2f bf = { ecol[4 * s], ecol[4 * s + 1] };   // ds_load_2addr_b64
            acc = __builtin_amdgcn_wmma_f32_16x16x4_f32(
                false, a[s], false, bf, (short)0, acc, false, false);
        }
        const int   c   = c0 + m;
        const float e2c = e2LDS[c];
        #pragma unroll
        for (int v = 0; v < 8; ++v) {
            const float score = __builtin_fmaf(-2.0f, acc[v], e2c);
            const bool  take  = score < mind[v];
            mind[v] = take ? score : mind[v];
            mini[v] = take ? c     : mini[v];
        }
    }

    // branchless min-reduce across the 16 lanes of each half (wave32 shuffles)
    #pragma unroll
    for (int off = 1; off < 16; off <<= 1) {
        #pragma unroll
        for (int v = 0; v < 8; ++v) {
            const float od = __shfl_xor(mind[v], off, 32);
            const int   oi = __shfl_xor(mini[v], off, 32);
            const bool take = (od < mind[v]) | ((od == mind[v]) & (oi < mini[v]));
            mind[v] = take ? od : mind[v];
            mini[v] = take ? oi : mini[v];
        }
    }
    if (m == 0) {   // lanes 0 and 16 publish rows 0..7 / 8..15
        #pragma unroll
        for (int v = 0; v < 8; ++v) {
            const int mm = 8 * half + v;
            idxLDS[wave * 16 + mm] = mini[v];
            idx_ws[nbase + mm]     = mini[v];
        }
    }
    __syncthreads();

    const int myidx = idxLDS[wave * 16 + m];

    // ---- vq_loss: (1+COMMIT) * mean((emb[idx]-x)^2), using resident A frags ----
    float lacc = 0.0f;
    {
        const float* eq = &embLDS[myidx * DIM + 2 * half];
        #pragma unroll
        for (int s = 0; s < 8; ++s) {
            const float e0 = eq[4 * s]     - a[s].x;
            const float e1 = eq[4 * s + 1] - a[s].y;
            lacc += e0 * e0 + e1 * e1;
        }
    }
    #pragma unroll
    for (int off = 1; off < 32; off <<= 1) lacc += __shfl_xor(lacc, off, 32);
    if (lane == 0)
        atomicAdd(loss, lacc * (1.26f / 33554432.0f));   // global_atomic_add_f32

    // ---- quantized output (straight-through == emb[idx]), [B,D,H,W], streamed ----
    {
        float* qb = qout + (long)b * DIM * HW + hw0 + m;
        const float* eq = &embLDS[myidx * DIM + 16 * half];
        #pragma unroll
        for (int t = 0; t < 16; ++t)
            __builtin_nontemporal_store(eq[t], qb + (long)(16 * half + t) * HW);
    }
}

// ---------------- kernel 3: scatter the ones into the one-hot ----------------
__global__ __launch_bounds__(256) void vq_scatter(const int* __restrict__ idx_ws,
                                                  int* __restrict__ disc) {
    const long n = (long)blockIdx.x * blockDim.x + threadIdx.x;   // < NVEC
    const int b  = (int)(n >> 14);
    const int hw = (int)(n & (HW - 1));
    const int c  = idx_ws[n];
    __builtin_nontemporal_store(1, disc + ((long)b * NCODE + c) * HW + hw);
}

// ---------------- launcher ----------------
extern "C" void kernel_launch(void* const* d_in, const int* in_sizes, int n_in,
                              void* d_out, int out_size, void* d_ws, size_t ws_size,
                              hipStream_t stream) {
    const float* x   = (const float*)d_in[0];   // [64,32,128,128]
    const float* emb = (const float*)d_in[1];   // [512,32]
    float* out   = (float*)d_out;
    float* qout  = out;                         // output 0
    float* loss  = out + LOSS_OFF;              // output 1 (scalar)
    int*   disc  = ((int*)d_out) + DISC_OFF;    // output 2 (int32 one-hot)
    int*   idxws = (int*)d_ws;                  // NVEC int32 scratch (4 MB)

    // 1) 2.15 GB zero-fill with NT b128 stores (+ zero loss slot)
    vq_fill<<<32768, 256, 0, stream>>>(out);
    // 2) WMMA distance/argmin core: 8192 blocks * 128 vectors = NVEC
    vq_main<<<8192, 256, 0, stream>>>(x, emb, qout, loss, idxws);
    // 3) scatter the single 1 per vector
    vq_scatter<<<(int)(NVEC / 256), 256, 0, stream>>>(idxws, disc);
}